// onering_conv_layer_batch_50543175139552
// MI455X (gfx1250) — compile-verified
//
#include <hip/hip_runtime.h>

typedef __attribute__((ext_vector_type(2))) float v2f;
typedef __attribute__((ext_vector_type(8))) float v8f;

#define C_IN   32
#define C_OUT  32
#define NNODE  163842
#define KTOT   224          // 7 * C_IN
#define LDSW_STRIDE 226     // padded row stride (dwords) -> conflict-free A reads

__global__ __launch_bounds__(256) void onering_conv_wmma_f32(
    const float* __restrict__ x,      // [B][C_IN][N]
    const float* __restrict__ W,      // [C_OUT][224]
    const float* __restrict__ bias,   // [C_OUT]
    const int*   __restrict__ neigh,  // [N][7]
    float*       __restrict__ out)    // [B][C_OUT][N]
{
    __shared__ __align__(16) float sW[C_OUT * LDSW_STRIDE];
    __shared__ __align__(16) float sB[C_OUT];

    const int tid = threadIdx.x;

    // Stage W (32x224) into LDS with padded stride; stage bias.
    for (int idx = tid; idx < C_OUT * KTOT; idx += 256) {
        int co = idx / KTOT;
        int k  = idx - co * KTOT;
        sW[co * LDSW_STRIDE + k] = W[idx];
    }
    if (tid < C_OUT) sB[tid] = bias[tid];
    __syncthreads();

    const int lane = tid & 31;
    const int wave = tid >> 5;
    const int lo   = lane & 15;   // N / M index within tile
    const int hi   = lane >> 4;   // K-half selector

    const int batch = blockIdx.y;
    const int n0    = blockIdx.x * 256 + wave * 32;   // 2 node-tiles per wave

    const float* xb = x   + (size_t)batch * C_IN  * NNODE;
    float*       ob = out + (size_t)batch * C_OUT * NNODE;
    const size_t sN = (size_t)NNODE;

    // Accumulators: [node-tile t][m-tile mt], preloaded with bias.
    // C/D layout: VGPR r holds (M = mt*16 + 8*hi + r, N = lo).
    v8f acc[2][2];
    #pragma unroll
    for (int r = 0; r < 8; ++r) {
        float b0 = sB[hi * 8 + r];
        float b1 = sB[16 + hi * 8 + r];
        acc[0][0][r] = b0; acc[1][0][r] = b0;
        acc[0][1][r] = b1; acc[1][1][r] = b1;
    }

    // Gather neighbor indices for the node each lane-column owns (clamped for tail).
    int nidx[2][7];
    #pragma unroll
    for (int t = 0; t < 2; ++t) {
        int n = n0 + t * 16 + lo;
        if (n > NNODE - 1) n = NNODE - 1;
        const int* nr = neigh + (size_t)n * 7;
        #pragma unroll
        for (int j = 0; j < 7; ++j) nidx[t][j] = nr[j];
    }

    for (int j = 0; j < 7; ++j) {
        // B operand column pointers: x[b][c][node], c = j-local channel.
        const float* p0 = xb + (size_t)nidx[0][j] + (size_t)(2 * hi) * sN;
        const float* p1 = xb + (size_t)nidx[1][j] + (size_t)(2 * hi) * sN;
        // A operand rows in LDS (K base = j*32 + 2*hi).
        const float* w0 = &sW[lo        * LDSW_STRIDE + j * 32 + 2 * hi];
        const float* w1 = &sW[(lo + 16) * LDSW_STRIDE + j * 32 + 2 * hi];

        #pragma unroll
        for (int s = 0; s < 8; ++s) {      // K step of 4: k0 = j*32 + 4*s
            v2f a0 = *(const v2f*)(w0 + s * 4);
            v2f a1 = *(const v2f*)(w1 + s * 4);
            v2f b0; b0.x = p0[0]; b0.y = p0[sN];
            v2f b1; b1.x = p1[0]; b1.y = p1[sN];
            p0 += 4 * sN; p1 += 4 * sN;

            acc[0][0] = __builtin_amdgcn_wmma_f32_16x16x4_f32(false, a0, false, b0, (short)0, acc[0][0], false, false);
            acc[0][1] = __builtin_amdgcn_wmma_f32_16x16x4_f32(false, a1, false, b0, (short)0, acc[0][1], false, false);
            acc[1][0] = __builtin_amdgcn_wmma_f32_16x16x4_f32(false, a0, false, b1, (short)0, acc[1][0], false, false);
            acc[1][1] = __builtin_amdgcn_wmma_f32_16x16x4_f32(false, a1, false, b1, (short)0, acc[1][1], false, false);
        }
    }

    // Store: out[b][co][n]; lanes 0-15 / 16-31 each write 64B-contiguous runs.
    #pragma unroll
    for (int t = 0; t < 2; ++t) {
        int n = n0 + t * 16 + lo;
        if (n < NNODE) {
            #pragma unroll
            for (int mt = 0; mt < 2; ++mt) {
                #pragma unroll
                for (int r = 0; r < 8; ++r) {
                    int co = mt * 16 + hi * 8 + r;
                    ob[(size_t)co * sN + n] = acc[t][mt][r];
                }
            }
        }
    }
}

extern "C" void kernel_launch(void* const* d_in, const int* in_sizes, int n_in,
                              void* d_out, int out_size, void* d_ws, size_t ws_size,
                              hipStream_t stream) {
    const float* x     = (const float*)d_in[0];
    const float* W     = (const float*)d_in[1];
    const float* bias  = (const float*)d_in[2];
    const int*   neigh = (const int*)d_in[3];
    float*       out   = (float*)d_out;

    int B = in_sizes[0] / (C_IN * NNODE);   // = 4
    dim3 grid((NNODE + 255) / 256, B, 1);
    onering_conv_wmma_f32<<<grid, dim3(256, 1, 1), 0, stream>>>(x, W, bias, neigh, out);
}